// SemiMarkovModule_79637283602642
// MI455X (gfx1250) — compile-verified
//
#include <hip/hip_runtime.h>
#include <hip/hip_bf16.h>
#include <math.h>

#define BIG_NEG (-1.0e9f)

typedef __attribute__((ext_vector_type(2))) float v2f;
typedef __attribute__((ext_vector_type(4))) float v4f;
typedef __attribute__((ext_vector_type(8))) float v8f;

// Fixed problem shape from the reference setup
#define B_    4
#define N1_   511
#define D_    64
#define C1_   31
#define K_    32
#define N_    512            // N1+1
#define C_    32             // C1+1
#define MROWS (B_*N1_)       // 2044
#define MPAD  2048
#define CSLEN (N_ + K_ + 1)  // 545

// workspace layout (float offsets; all multiples of 4 for float4 access)
#define WS_TRANS 0u          // [32][32]
#define WS_INIT  1024u       // [32]
#define WS_LEN   1056u       // [32][32]
#define WS_BIAS  2080u       // [32]
#define WS_BMAT  2112u       // [64][96]  = [ P | Pm(31 cols) | 0 ]
#define WS_EMIS  8256u       // [4][512][32]
#define WS_CS    73792u      // [4][545][32]

// ---------------------------------------------------------------------------
// Kernel 1: setup — Cholesky, Linv, P, Bmat, bias, trans_a, init_a, length_a,
// EOS emission row. One block of 64 threads; all heavy state kept in LDS.
// ---------------------------------------------------------------------------
__global__ __launch_bounds__(64) void smm_setup(
    const float* __restrict__ tlog, const float* __restrict__ ilog,
    const float* __restrict__ plr,  const float* __restrict__ means,
    const float* __restrict__ cov,  float* __restrict__ ws)
{
    __shared__ float sL[64*64];
    __shared__ float sLi[64*64];
    __shared__ float sP[64*64];
    __shared__ float s_logdet;
    const int tid = threadIdx.x;

    for (int i = tid; i < 4096; i += 64) sL[i] = cov[i];
    __syncthreads();

    // In-place Cholesky (lower), serial over columns.
    for (int j = 0; j < 64; ++j) {
        if (tid == j) {
            float s = sL[j*64+j];
            for (int m = 0; m < j; ++m) s -= sL[j*64+m]*sL[j*64+m];
            sL[j*64+j] = sqrtf(s);
        }
        __syncthreads();
        if (tid > j) {
            float s = sL[tid*64+j];
            for (int m = 0; m < j; ++m) s -= sL[tid*64+m]*sL[j*64+m];
            sL[tid*64+j] = s / sL[j*64+j];
        }
        __syncthreads();
    }

    // Linv: thread j solves L x = e_j (forward substitution), column j.
    {
        const int j = tid;
        for (int i = 0; i < 64; ++i) {
            float x = 0.f;
            if (i >= j) {
                float s = (i == j) ? 1.f : 0.f;
                for (int m = j; m < i; ++m) s -= sL[i*64+m]*sLi[m*64+j];
                x = s / sL[i*64+i];
            }
            sLi[i*64+j] = x;
        }
    }
    __syncthreads();

    // P = Linv^T Linv (covariance inverse).
    for (int idx = tid; idx < 4096; idx += 64) {
        const int d0 = idx >> 6, e = idx & 63;
        const int m0 = d0 > e ? d0 : e;
        float s = 0.f;
        for (int m = m0; m < 64; ++m) s += sLi[m*64+d0]*sLi[m*64+e];
        sP[idx] = s;
    }
    __syncthreads();

    // Bmat row d: [ P[d,0..63] | (P m_c)[d] for c<31 | 0 ]
    {
        const int d0 = tid;
        float* bm = ws + WS_BMAT + d0*96;
        for (int e = 0; e < 64; ++e) bm[e] = sP[d0*64+e];
        for (int c = 0; c < C1_; ++c) {
            float s = 0.f;
            for (int e = 0; e < 64; ++e) s += sP[d0*64+e]*means[c*64+e];
            bm[64+c] = s;
        }
        bm[95] = 0.f;
    }

    if (tid == 0) {
        float ld = 0.f;
        for (int i = 0; i < 64; ++i) ld += logf(sL[i*64+i]);
        s_logdet = 2.f*ld;
    }
    __syncthreads();

    // bias[c] = -0.5*(d*log(2pi) + logdet + m_c^T P m_c)
    if (tid < 32) {
        float v = 0.f;
        if (tid < C1_) {
            float q = 0.f;
            for (int d0 = 0; d0 < 64; ++d0) {
                float s = 0.f;
                for (int e = 0; e < 64; ++e) s += sP[d0*64+e]*means[tid*64+e];
                q += means[tid*64+d0]*s;
            }
            v = -0.5f*(64.f*logf(6.2831853071795864f) + s_logdet + q);
        }
        ws[WS_BIAS + tid] = v;
    }

    // trans_a defaults: row 31 -> 0, col 31 (rows<31) -> BIG_NEG (non-overlapping)
    for (int idx = tid; idx < 1024; idx += 64) {
        const int i = idx >> 5, j = idx & 31;
        if (i == 31)      ws[WS_TRANS + idx] = 0.f;
        else if (j == 31) ws[WS_TRANS + idx] = BIG_NEG;
    }
    // trans_a[:31,:31] = log_softmax over axis 0 of diag-masked logits; thread j = column j
    if (tid < C1_) {
        const int j = tid;
        float mx = -3.0e38f;
        for (int i = 0; i < C1_; ++i) {
            const float v = (i == j) ? BIG_NEG : tlog[i*C1_ + j];
            mx = fmaxf(mx, v);
        }
        float se = 0.f;
        for (int i = 0; i < C1_; ++i) {
            const float v = (i == j) ? BIG_NEG : tlog[i*C1_ + j];
            se += expf(v - mx);
        }
        const float lse = mx + logf(se);
        for (int i = 0; i < C1_; ++i) {
            const float v = (i == j) ? BIG_NEG : tlog[i*C1_ + j];
            ws[WS_TRANS + i*32 + j] = v - lse;
        }
    }

    // init_a
    if (tid == 0) {
        float mx = -3.0e38f;
        for (int i = 0; i < C1_; ++i) mx = fmaxf(mx, ilog[i]);
        float se = 0.f;
        for (int i = 0; i < C1_; ++i) se += expf(ilog[i]-mx);
        const float lse = mx + logf(se);
        for (int i = 0; i < C1_; ++i) ws[WS_INIT + i] = ilog[i] - lse;
        ws[WS_INIT + 31] = BIG_NEG;
    }

    // length_a[k][c]
    for (int idx = tid; idx < 1024; idx += 64) {
        const int k = idx >> 5, c = idx & 31;
        float v;
        if (c < C1_) {
            const float p = plr[c];
            v = (float)k * p - expf(p) - lgammaf((float)k + 1.f);
        } else {
            v = (k == 1) ? 0.f : BIG_NEG;
        }
        ws[WS_LEN + idx] = v;
    }

    // EOS emission rows: emis_a[b][N1][c] = (c==31 ? 0 : BIG_NEG)
    for (int idx = tid; idx < B_*32; idx += 64) {
        const int b = idx >> 5, c = idx & 31;
        ws[WS_EMIS + (unsigned)((b*N_ + N1_)*32 + c)] = (c == 31) ? 0.f : BIG_NEG;
    }
}

// ---------------------------------------------------------------------------
// Kernel 2: emission log-probs via fp32 WMMA.
// One wave per 16-row tile of F=[2048,64]; GEMM F x Bmat[64,96] using
// v_wmma_f32_16x16x4_f32.  Cols 0..63 give G=FP (row-dot with F -> f^T P f),
// cols 64..95 give f.Pm_c.  emis = (f.Pm_c) - 0.5 f^T P f + bias[c].
// ---------------------------------------------------------------------------
__global__ __launch_bounds__(256) void smm_emis(
    const float* __restrict__ feat, float* __restrict__ ws)
{
    __shared__ float sB[64*96];
    for (int i = threadIdx.x; i < 64*96; i += 256) sB[i] = ws[WS_BMAT + i];
    __syncthreads();

    const int lane = threadIdx.x & 31;
    const int cl   = lane & 15;     // column within 16 / A-row within 16
    const int hi   = lane >> 4;     // half-wave select
    const int tile = blockIdx.x*8 + (threadIdx.x >> 5);
    const int rowBase = tile*16;

    const int   mA     = rowBase + cl;
    const int   mAc    = (mA < MROWS) ? mA : 0;
    const float aScale = (mA < MROWS) ? 1.f : 0.f;

    v8f acc0 = {}, acc1 = {}, acc2 = {}, acc3 = {}, acc4 = {}, acc5 = {};

#define WMMA_STEP(ACC, CT)                                                    \
    do {                                                                      \
        v2f bf;                                                               \
        bf.x = sB[k0*96 + (CT)*16 + cl];                                      \
        bf.y = sB[(k0+1)*96 + (CT)*16 + cl];                                  \
        ACC = __builtin_amdgcn_wmma_f32_16x16x4_f32(                          \
            false, a, false, bf, (short)0, ACC, false, false);                \
    } while (0)

#pragma unroll
    for (int k = 0; k < 64; k += 4) {
        const int k0 = k + hi*2;   // lanes 0-15: K=k,k+1 ; lanes 16-31: K=k+2,k+3
        v2f a;
        a.x = feat[mAc*64 + k0]     * aScale;
        a.y = feat[mAc*64 + k0 + 1] * aScale;
        WMMA_STEP(acc0, 0);
        WMMA_STEP(acc1, 1);
        WMMA_STEP(acc2, 2);
        WMMA_STEP(acc3, 3);
        WMMA_STEP(acc4, 4);
        WMMA_STEP(acc5, 5);
    }
#undef WMMA_STEP

    // f^T P f : row-dot of G (acc0..3) with F, then 16-lane xor reduction.
    float fpf[8];
#pragma unroll
    for (int r = 0; r < 8; ++r) {
        const int m  = rowBase + r + 8*hi;
        const int mc = (m < MROWS) ? m : 0;
        float s = acc0[r]*feat[mc*64 +  0 + cl]
                + acc1[r]*feat[mc*64 + 16 + cl]
                + acc2[r]*feat[mc*64 + 32 + cl]
                + acc3[r]*feat[mc*64 + 48 + cl];
        s += __shfl_xor(s, 1, 16);
        s += __shfl_xor(s, 2, 16);
        s += __shfl_xor(s, 4, 16);
        s += __shfl_xor(s, 8, 16);
        fpf[r] = s;
    }

    const float* bias = ws + WS_BIAS;
#pragma unroll
    for (int half = 0; half < 2; ++half) {
        const int c  = half*16 + cl;
        const float bc = bias[c];
#pragma unroll
        for (int r = 0; r < 8; ++r) {
            const int m = rowBase + r + 8*hi;
            float v = (half == 0 ? acc4[r] : acc5[r]) - 0.5f*fpf[r] + bc;
            if (c == 31) v = BIG_NEG;
            if (m < MROWS) {
                const int bb = m / N1_;
                const int n  = m - bb*N1_;
                ws[WS_EMIS + (unsigned)((bb*N_ + n)*32 + c)] = v;
            }
        }
    }
}

// ---------------------------------------------------------------------------
// Kernel 3: time cumsum of padded emissions. One block (32 lanes = classes)
// per batch element; tiny (4*545*32 values).
// ---------------------------------------------------------------------------
__global__ __launch_bounds__(32) void smm_cumsum(float* __restrict__ ws)
{
    const int b = blockIdx.x;
    const int c = threadIdx.x;
    float acc = 0.f;
    ws[WS_CS + (unsigned)((b*CSLEN + 0)*32 + c)] = 0.f;
#pragma unroll 8
    for (int j = 0; j < CSLEN-1; ++j) {
        const float e = (j < N_) ? ws[WS_EMIS + (unsigned)((b*N_ + j)*32 + c)] : 0.f;
        acc += e;
        ws[WS_CS + (unsigned)((b*CSLEN + j + 1)*32 + c)] = acc;
    }
}

// ---------------------------------------------------------------------------
// Kernel 4: the 268 MB writer.  One wave per (b,t,k) tile of 32x32 = 4 KB.
// trans_a staged into LDS with global_load_async_to_lds_b128 (ASYNCcnt path);
// stores are coalesced non-temporal b128.
// ---------------------------------------------------------------------------
__global__ __launch_bounds__(256) void smm_scores(
    const float* __restrict__ ws, float* __restrict__ out)
{
    __shared__ float s_trans[1024];

    // Async-copy the 4 KB trans_a table into LDS: 256 threads x 16 B.
    {
        const unsigned lds_addr = (unsigned)(size_t)(&s_trans[0]) + threadIdx.x*16u;
        const unsigned goff     = threadIdx.x * 16u;
        const float* gsrc = ws + WS_TRANS;
        asm volatile("global_load_async_to_lds_b128 %0, %1, %2"
                     :: "v"(lds_addr), "v"(goff), "s"(gsrc)
                     : "memory");
        asm volatile("s_wait_asynccnt 0" ::: "memory");
    }
    __syncthreads();

    const int wid  = blockIdx.x*8 + (threadIdx.x >> 5);   // 0 .. 65407
    const int lane = threadIdx.x & 31;
    const int k = wid & 31;
    const int t = (wid >> 5) % N1_;
    const int b = wid / (N1_*32);

    const int q = lane & 7;    // v4f index along c (c = 4q..4q+3)
    const int p = lane >> 3;   // cp sub-row within a 4-row group

    const v4f* cs4   = (const v4f*)(ws + WS_CS);
    const v4f* len4p = (const v4f*)(ws + WS_LEN);
    const v4f* ini4p = (const v4f*)(ws + WS_INIT);

    const v4f ca = cs4[(b*CSLEN + t + k)*8 + q];
    const v4f cb = cs4[(b*CSLEN + t)*8 + q];
    const v4f l4 = len4p[k*8 + q];

    v4f base = ca - cb + l4;
    if (t == 0) {
        base += ini4p[q];
    }
    const bool eos = (k >= 1) && (t == N1_ - k);   // t == N-1-k

    const v4f* st4 = (const v4f*)s_trans;
    v4f* outp = (v4f*)out + (size_t)wid*256 + lane;

#pragma unroll
    for (int j = 0; j < 8; ++j) {
        const int cp = j*4 + p;
        const v4f tr = st4[cp*8 + q];
        const float add = (eos && cp < 31) ? BIG_NEG : 0.f;
        v4f v = base + tr + add;
        __builtin_nontemporal_store(v, outp + j*32);
    }
}

// ---------------------------------------------------------------------------
extern "C" void kernel_launch(void* const* d_in, const int* in_sizes, int n_in,
                              void* d_out, int out_size, void* d_ws, size_t ws_size,
                              hipStream_t stream) {
    const float* features = (const float*)d_in[0];
    const float* tlogits  = (const float*)d_in[1];
    const float* ilogits  = (const float*)d_in[2];
    const float* plrates  = (const float*)d_in[3];
    const float* means    = (const float*)d_in[4];
    const float* cov      = (const float*)d_in[5];
    float* ws  = (float*)d_ws;
    float* out = (float*)d_out;

    smm_setup<<<1, 64, 0, stream>>>(tlogits, ilogits, plrates, means, cov, ws);
    smm_emis<<<MPAD/16/8, 256, 0, stream>>>(features, ws);      // 16 blocks, 128 waves
    smm_cumsum<<<B_, 32, 0, stream>>>(ws);
    smm_scores<<<(B_*N1_*K_)/8, 256, 0, stream>>>(ws, out);     // 8176 blocks
}